// GATLayer_19739669692891
// MI455X (gfx1250) — compile-verified
//
#include <hip/hip_runtime.h>
#include <math.h>

#define H 8
#define F_IN 256
#define F_OUT 32
#define C_TOT (H * F_OUT)  // 256 = concat width, == F_IN (residual is identity)
#define KPANEL 32          // K-elements staged per LDS panel
#define PANEL_FLOATS (H * KPANEL * F_OUT)  // 8192 floats = 32KB

typedef float v2f __attribute__((ext_vector_type(2)));
typedef float v8f __attribute__((ext_vector_type(8)));
typedef unsigned int v4u __attribute__((ext_vector_type(4)));
typedef int v4i __attribute__((ext_vector_type(4)));
typedef int v8i __attribute__((ext_vector_type(8)));

#if defined(__has_builtin)
#if __has_builtin(__builtin_amdgcn_tensor_load_to_lds) && \
    __has_builtin(__builtin_amdgcn_s_wait_tensorcnt)
#define HAVE_TDM 1
#endif
#endif

// ---------------------------------------------------------------------------
// K0: seed residual (out = x) and zero the softmax denominators.
// ---------------------------------------------------------------------------
__global__ void gat_init_kernel(const float* __restrict__ x,
                                float* __restrict__ out,
                                float* __restrict__ attn_sum, int N) {
  int tid = blockIdx.x * blockDim.x + threadIdx.x;
  int total4 = N * (C_TOT / 4);
  if (tid < total4) {
    reinterpret_cast<float4*>(out)[tid] =
        reinterpret_cast<const float4*>(x)[tid];
  }
  if (tid < N * H) attn_sum[tid] = 0.0f;
}

#ifdef HAVE_TDM
// Issue a TDM 2D tile load: W panel (8 heads x 32 K x 32 cols, fp32) -> LDS.
//   tile_dim0 = 1024 elements (one head's 32x32 K-panel, contiguous in W)
//   tile_dim1 = 8 heads, dim0-stride = 8192 elements (head slab stride)
// D# layout per CDNA5 ISA.
__device__ __forceinline__ void tdm_load_panel(const float* gsrc,
                                               unsigned int lds_byte_addr) {
  unsigned long long ga = (unsigned long long)(uintptr_t)gsrc;
  v4u g0;
  g0.x = 1u;                                  // count=1, user descriptor
  g0.y = lds_byte_addr;                       // lds_addr (bytes)
  g0.z = (unsigned int)ga;                    // global_addr[31:0]
  g0.w = (unsigned int)((ga >> 32) & 0x1FFFFFFu) | (2u << 30);  // type=2
  v8i g1;
  g1[0] = 0x20000;                 // workgroup_mask=0, data_size=2 (4 bytes)
  g1[1] = (int)(0x2000u << 16);    // tensor_dim0 = 8192 (low 16 -> [31:16])
  g1[2] = (int)(8u << 16);         // tensor_dim0 hi=0 | tensor_dim1 = 8
  g1[3] = (int)(1024u << 16);      // tensor_dim1 hi=0 | tile_dim0 = 1024
  g1[4] = 8;                       // tile_dim1 = 8, tile_dim2 = 0
  g1[5] = 8192;                    // tensor_dim0_stride[31:0] = 8192 elems
  g1[6] = 0;                       // stride hi, tensor_dim1_stride lo
  g1[7] = 0;
  v4i z4 = {};
#if __clang_major__ >= 23
  v8i z8 = {};
  __builtin_amdgcn_tensor_load_to_lds(g0, g1, z4, z4, z8, 0);
#else
  __builtin_amdgcn_tensor_load_to_lds(g0, g1, z4, z4, 0);
#endif
}
#endif

// ---------------------------------------------------------------------------
// K1: h = x @ W  via V_WMMA_F32_16X16X4_F32 (fp32 in / fp32 acc).
// Block = 256 threads = 8 waves; block owns 16 rows x 256 cols.
// Wave w owns head w (two 16x16 col tiles). W is staged in LDS in K=32
// panels, double-buffered, prefetched by the Tensor Data Mover while WMMAs
// consume the previous panel (B reads become ds_load instead of thrashing
// the 64KB WGP$ with 256KB of W).
// ---------------------------------------------------------------------------
__global__ __launch_bounds__(256) void gat_proj_wmma_kernel(
    const float* __restrict__ x, const float* __restrict__ W,
    float* __restrict__ hbuf, int N) {
  __shared__ float ws[2][PANEL_FLOATS];  // 2 x 32KB panels

  const int wave = threadIdx.x >> 5;  // 0..7 == head index
  const int lane = threadIdx.x & 31;
  const int rowBase = blockIdx.x * 16;
  const int lrow = lane & 15;
  const int khalf = (lane >> 4) << 1;  // 0 for lanes 0-15, 2 for lanes 16-31

  int arow = rowBase + lrow;
  if (arow >= N) arow = N - 1;  // clamp (N is a multiple of 16 in practice)
  const float* xrow = x + (size_t)arow * F_IN;

  v8f c0 = {};
  v8f c1 = {};
  const int nPanels = F_IN / KPANEL;  // 8

#ifdef HAVE_TDM
  const unsigned int ldsBase = (unsigned int)(uintptr_t)(&ws[0][0]);
  if (wave == 0) {
    tdm_load_panel(W, ldsBase);
    __builtin_amdgcn_s_wait_tensorcnt(0);
  }
  __syncthreads();
  for (int kp = 0; kp < nPanels; ++kp) {
    const int buf = kp & 1;
    if (wave == 0 && kp + 1 < nPanels) {  // prefetch next panel via TDM
      tdm_load_panel(W + (size_t)(kp + 1) * (KPANEL * F_OUT),
                     ldsBase + (unsigned int)((buf ^ 1) * sizeof(float) *
                                              PANEL_FLOATS));
    }
    const float* xp = xrow + kp * KPANEL + khalf;
#pragma unroll
    for (int kk = 0; kk < KPANEL; kk += 4) {
      v2f a = *reinterpret_cast<const v2f*>(xp + kk);
      const float* wsp = &ws[buf][(wave * KPANEL + kk + khalf) * F_OUT + lrow];
      v2f b0, b1;
      b0.x = wsp[0];
      b0.y = wsp[F_OUT];
      b1.x = wsp[16];
      b1.y = wsp[F_OUT + 16];
      c0 = __builtin_amdgcn_wmma_f32_16x16x4_f32(false, a, false, b0, (short)0,
                                                 c0, false, false);
      c1 = __builtin_amdgcn_wmma_f32_16x16x4_f32(false, a, false, b1, (short)0,
                                                 c1, false, false);
    }
    __syncthreads();  // all waves done reading ws[buf]
    if (wave == 0 && kp + 1 < nPanels) {
      __builtin_amdgcn_s_wait_tensorcnt(0);  // next panel landed in LDS
    }
    __syncthreads();
  }
#else
  // Fallback: cooperative vectorized panel copy (single buffer).
  const int chunk = threadIdx.x >> 5;  // head chunk this thread copies
  const int cl = threadIdx.x & 31;
  for (int kp = 0; kp < nPanels; ++kp) {
    __syncthreads();
    const float4* src = reinterpret_cast<const float4*>(
        W + (size_t)chunk * (F_IN * F_OUT) + (size_t)kp * (KPANEL * F_OUT));
    float4* dst = reinterpret_cast<float4*>(&ws[0][chunk * (KPANEL * F_OUT)]);
#pragma unroll
    for (int j = 0; j < 8; ++j) dst[cl + j * 32] = src[cl + j * 32];
    __syncthreads();
    const float* xp = xrow + kp * KPANEL + khalf;
#pragma unroll
    for (int kk = 0; kk < KPANEL; kk += 4) {
      v2f a = *reinterpret_cast<const v2f*>(xp + kk);
      const float* wsp = &ws[0][(wave * KPANEL + kk + khalf) * F_OUT + lrow];
      v2f b0, b1;
      b0.x = wsp[0];
      b0.y = wsp[F_OUT];
      b1.x = wsp[16];
      b1.y = wsp[F_OUT + 16];
      c0 = __builtin_amdgcn_wmma_f32_16x16x4_f32(false, a, false, b0, (short)0,
                                                 c0, false, false);
      c1 = __builtin_amdgcn_wmma_f32_16x16x4_f32(false, a, false, b1, (short)0,
                                                 c1, false, false);
    }
  }
#endif

  float* outBase = hbuf + (size_t)rowBase * C_TOT + wave * F_OUT;
  const int rhi = (lane >> 4) << 3;  // 0 or 8
#pragma unroll
  for (int v = 0; v < 8; ++v) {
    int row = rhi + v;
    if (rowBase + row < N) {
      outBase[(size_t)row * C_TOT + lrow] = c0[v];
      outBase[(size_t)row * C_TOT + 16 + lrow] = c1[v];
    }
  }
}

// ---------------------------------------------------------------------------
// K2: attn_src[n,h] = <h[n,h,:], a_src[h,:]>, same for dst. 1 thread/(n,h).
// ---------------------------------------------------------------------------
__global__ void gat_attn_dot_kernel(const float* __restrict__ hbuf,
                                    const float* __restrict__ a_src,
                                    const float* __restrict__ a_dst,
                                    float* __restrict__ attn_src,
                                    float* __restrict__ attn_dst, int N) {
  int tid = blockIdx.x * blockDim.x + threadIdx.x;  // n*H + head
  if (tid >= N * H) return;
  int n = tid >> 3, head = tid & 7;
  const float4* hv =
      reinterpret_cast<const float4*>(hbuf + (size_t)n * C_TOT + head * F_OUT);
  const float4* as = reinterpret_cast<const float4*>(a_src + head * F_OUT);
  const float4* ad = reinterpret_cast<const float4*>(a_dst + head * F_OUT);
  float s = 0.0f, d = 0.0f;
#pragma unroll
  for (int j = 0; j < 8; ++j) {
    float4 h4 = hv[j], a4 = as[j], b4 = ad[j];
    s += h4.x * a4.x + h4.y * a4.y + h4.z * a4.z + h4.w * a4.w;
    d += h4.x * b4.x + h4.y * b4.y + h4.z * b4.z + h4.w * b4.w;
  }
  attn_src[tid] = s;
  attn_dst[tid] = d;
}

__device__ __forceinline__ float leaky_exp(float v) {
  v = v > 0.0f ? v : 0.2f * v;  // leaky_relu, slope 0.2
  return __expf(v);             // global-max shift cancels in the softmax
}

// ---------------------------------------------------------------------------
// K3: softmax denominators: attn_sum[dst,h] += exp(lrelu(as[src]+ad[dst])).
// ---------------------------------------------------------------------------
__global__ void gat_edge_sum_kernel(const int* __restrict__ ei,
                                    const float* __restrict__ attn_src,
                                    const float* __restrict__ attn_dst,
                                    float* __restrict__ attn_sum, int E) {
  int e = blockIdx.x * blockDim.x + threadIdx.x;
  if (e >= E) return;
  int s = ei[e], d = ei[E + e];
#pragma unroll
  for (int h = 0; h < H; ++h) {
    float p = leaky_exp(attn_src[s * H + h] + attn_dst[d * H + h]);
    unsafeAtomicAdd(&attn_sum[d * H + h], p);
  }
}

// ---------------------------------------------------------------------------
// K4: out[dst] += (exp/sum) * h[src]. One wave32 per edge; lane L handles
// 8 consecutive floats (all inside head L>>2). ~L2-resident gather+atomics.
// ---------------------------------------------------------------------------
__global__ __launch_bounds__(256) void gat_edge_aggregate_kernel(
    const int* __restrict__ ei, const float* __restrict__ hbuf,
    const float* __restrict__ attn_src, const float* __restrict__ attn_dst,
    const float* __restrict__ attn_sum, float* __restrict__ out, int E) {
  int wid = (blockIdx.x * blockDim.x + threadIdx.x) >> 5;
  int lane = threadIdx.x & 31;
  if (wid >= E) return;
  int s = ei[wid], d = ei[E + wid];
  int head = lane >> 2;

  float p = leaky_exp(attn_src[s * H + head] + attn_dst[d * H + head]);
  float w = p / (attn_sum[d * H + head] + 1e-10f);

  const float4* hp =
      reinterpret_cast<const float4*>(hbuf + (size_t)s * C_TOT + lane * 8);
  float4 h0 = hp[0], h1 = hp[1];
  float* op = out + (size_t)d * C_TOT + lane * 8;
  unsafeAtomicAdd(op + 0, w * h0.x);
  unsafeAtomicAdd(op + 1, w * h0.y);
  unsafeAtomicAdd(op + 2, w * h0.z);
  unsafeAtomicAdd(op + 3, w * h0.w);
  unsafeAtomicAdd(op + 4, w * h1.x);
  unsafeAtomicAdd(op + 5, w * h1.y);
  unsafeAtomicAdd(op + 6, w * h1.z);
  unsafeAtomicAdd(op + 7, w * h1.w);
}

// ---------------------------------------------------------------------------
extern "C" void kernel_launch(void* const* d_in, const int* in_sizes, int n_in,
                              void* d_out, int out_size, void* d_ws,
                              size_t ws_size, hipStream_t stream) {
  const float* x = (const float*)d_in[0];
  const int* ei = (const int*)d_in[1];  // [2, E]: row 0 = src, row 1 = dst
  const float* W = (const float*)d_in[2];
  const float* a_src = (const float*)d_in[3];
  const float* a_dst = (const float*)d_in[4];

  const int N = in_sizes[0] / F_IN;
  const int E = in_sizes[1] / 2;
  float* out = (float*)d_out;

  // Workspace: hbuf (N*256) | attn_src (N*8) | attn_dst (N*8) | attn_sum (N*8)
  float* hbuf = (float*)d_ws;
  float* w_attn_src = hbuf + (size_t)N * C_TOT;
  float* w_attn_dst = w_attn_src + (size_t)N * H;
  float* w_attn_sum = w_attn_dst + (size_t)N * H;

  int t_init = N * (C_TOT / 4);
  gat_init_kernel<<<(t_init + 255) / 256, 256, 0, stream>>>(x, out, w_attn_sum,
                                                            N);
  gat_proj_wmma_kernel<<<(N + 15) / 16, 256, 0, stream>>>(x, W, hbuf, N);
  gat_attn_dot_kernel<<<(N * H + 255) / 256, 256, 0, stream>>>(
      hbuf, a_src, a_dst, w_attn_src, w_attn_dst, N);
  gat_edge_sum_kernel<<<(E + 255) / 256, 256, 0, stream>>>(
      ei, w_attn_src, w_attn_dst, w_attn_sum, E);
  long long aggThreads = (long long)E * 32;
  gat_edge_aggregate_kernel<<<(int)((aggThreads + 255) / 256), 256, 0,
                              stream>>>(ei, hbuf, w_attn_src, w_attn_dst,
                                        w_attn_sum, out, E);
}